// EGAT_53635551592805
// MI455X (gfx1250) — compile-verified
//
#include <hip/hip_runtime.h>
#include <math.h>

// ---------------------------------------------------------------------------
// EGAT on MI455X (gfx1250, wave32).
// - Edge MLP fused per 16-edge tile in LDS using fp32 WMMA (16x16x4); only
//   a_edge[E,H] and the dst-segment-sum loop_sum[N,128] reach HBM (the 410MB
//   [E,128] intermediate never does).
// - Node transform xp = x @ W_x via WMMA.
// - Segment softmax via float atomic-max + atomic-add; softmax denominator is
//   applied once per (node,head) in the finalize kernel (scalar factored out
//   of the edge sum), so the 25.6M-thread aggregation does gather+FMA+atomic
//   only, vectorized float4.
// ---------------------------------------------------------------------------

#define NN 50000
#define NE 800000
#define DHID 128
#define DED 32

typedef float v2f __attribute__((ext_vector_type(2)));
typedef float v8f __attribute__((ext_vector_type(8)));

__device__ __forceinline__ v8f wmma4(v2f a, v2f b, v8f c) {
  // V_WMMA_F32_16X16X4_F32 : D = A(16x4) x B(4x16) + C(16x16), fp32 exact
  return __builtin_amdgcn_wmma_f32_16x16x4_f32(false, a, false, b, (short)0, c,
                                               false, false);
}

__device__ __forceinline__ float leaky02(float x) {
  return x > 0.f ? x : 0.2f * x;
}

__device__ __forceinline__ void atomicMaxF(float* addr, float v) {
  if (v >= 0.f)
    atomicMax((int*)addr, __float_as_int(v));
  else
    atomicMin((unsigned int*)addr, __float_as_uint(v));
}

// ------------------------------------------------------------------ fill ---
__global__ void k_fill(float* __restrict__ p, int n, float v) {
  int i = blockIdx.x * blockDim.x + threadIdx.x;
  if (i < n) p[i] = v;
}

// V[k][h] = sum_c W_e[k][h*C+c] * att_edge[h][c]   (128 x H, tiny)
__global__ void k_prep_vedge(const float* __restrict__ W_e,
                             const float* __restrict__ att_edge,
                             float* __restrict__ V, int H, int C, int HC) {
  int k = threadIdx.x;  // 128 threads
  for (int h = 0; h < H; ++h) {
    float acc = 0.f;
    for (int c = 0; c < C; ++c) acc += W_e[k * HC + h * C + c] * att_edge[h * C + c];
    V[k * H + h] = acc;
  }
}

// ------------------------------------------------- xp = X[N,128] @ W[128,HC]
__global__ void k_gemm_node(const float* __restrict__ X,
                            const float* __restrict__ W,
                            float* __restrict__ Y, int HC) {
  __shared__ float sA[16 * 128];
  const int t = threadIdx.x;
  const int mbase = blockIdx.x * 16;
  for (int i = t; i < 16 * 128; i += 256) sA[i] = X[(size_t)mbase * 128 + i];
  __syncthreads();
  const int wv = t >> 5;
  if (wv * 16 < HC) {
    const int lane = t & 31;
    const int r16 = lane & 15;        // A row / B col within tile
    const int kq = (lane >> 4) << 1;  // 0 or 2 : K sub-offset
    const int colg = wv * 16 + r16;
    v8f acc = {};
    for (int k0 = 0; k0 < 128; k0 += 4) {
      v2f a, b;
      a.x = sA[r16 * 128 + k0 + kq];
      a.y = sA[r16 * 128 + k0 + kq + 1];
      b.x = W[(k0 + kq) * HC + colg];
      b.y = W[(k0 + kq + 1) * HC + colg];
      acc = wmma4(a, b, acc);
    }
    const int mrow = mbase + 8 * (lane >> 4);
    for (int r = 0; r < 8; ++r) Y[(size_t)(mrow + r) * HC + colg] = acc[r];
  }
}

// a_src[n,h] = xp[n,h*C:] . att_src[h,:]   (and same for dst)
__global__ void k_att_nodes(const float* __restrict__ xp,
                            const float* __restrict__ att_src,
                            const float* __restrict__ att_dst,
                            float* __restrict__ a_src, float* __restrict__ a_dst,
                            int H, int C, int HC) {
  int i = blockIdx.x * blockDim.x + threadIdx.x;
  if (i >= NN * H) return;
  int n = i / H, h = i % H;
  const float* row = xp + (size_t)n * HC + h * C;
  float as = 0.f, ad = 0.f;
  for (int c = 0; c < C; ++c) {
    as += row[c] * att_src[h * C + c];
    ad += row[c] * att_dst[h * C + c];
  }
  a_src[i] = as;
  a_dst[i] = ad;
}

// -------- fused edge MLP: ReLU(EA@We1+be1)@We2+be2 per 16-edge tile in LDS,
// -------- emit a_edge = e@V, scatter e into loop_sum[dst], count deg[dst].
__global__ void k_edge_fused(const float* __restrict__ EA,
                             const int* __restrict__ ei,
                             const float* __restrict__ We1,
                             const float* __restrict__ be1,
                             const float* __restrict__ We2,
                             const float* __restrict__ be2,
                             const float* __restrict__ V,
                             float* __restrict__ a_edge,
                             float* __restrict__ loop_sum,
                             float* __restrict__ deg, int H) {
  __shared__ float sEA[16 * DED];
  __shared__ float sT1[16 * 128];
  __shared__ float sE[16 * 128];
  __shared__ int sDst[16];
  const int t = threadIdx.x;
  const size_t ebase = (size_t)blockIdx.x * 16;
  for (int i = t; i < 16 * DED; i += 256) sEA[i] = EA[ebase * DED + i];
  if (t < 16) sDst[t] = ei[NE + ebase + t];
  __syncthreads();

  const int wv = t >> 5, lane = t & 31;
  const int r16 = lane & 15;
  const int kq = (lane >> 4) << 1;
  const int colg = wv * 16 + r16;
  const int mhalf = 8 * (lane >> 4);

  {  // GEMM1: [16,32] @ [32,128] -> ReLU(+be1) -> sT1
    v8f acc = {};
    for (int k0 = 0; k0 < DED; k0 += 4) {
      v2f a, b;
      a.x = sEA[r16 * DED + k0 + kq];
      a.y = sEA[r16 * DED + k0 + kq + 1];
      b.x = We1[(k0 + kq) * 128 + colg];
      b.y = We1[(k0 + kq + 1) * 128 + colg];
      acc = wmma4(a, b, acc);
    }
    float bb = be1[colg];
    for (int r = 0; r < 8; ++r) {
      float v0 = acc[r] + bb;
      sT1[(mhalf + r) * 128 + colg] = v0 > 0.f ? v0 : 0.f;
    }
  }
  __syncthreads();
  {  // GEMM2: [16,128] @ [128,128] -> +be2 -> sE
    v8f acc = {};
    for (int k0 = 0; k0 < 128; k0 += 4) {
      v2f a, b;
      a.x = sT1[r16 * 128 + k0 + kq];
      a.y = sT1[r16 * 128 + k0 + kq + 1];
      b.x = We2[(k0 + kq) * 128 + colg];
      b.y = We2[(k0 + kq + 1) * 128 + colg];
      acc = wmma4(a, b, acc);
    }
    float bb = be2[colg];
    for (int r = 0; r < 8; ++r) sE[(mhalf + r) * 128 + colg] = acc[r] + bb;
  }
  __syncthreads();

  if (t < 16 * H) {  // a_edge[e,h] = e_row . V[:,h]
    int r = t / H, h = t % H;
    float acc = 0.f;
    for (int k = 0; k < 128; ++k) acc += sE[r * 128 + k] * V[k * H + h];
    a_edge[(ebase + r) * H + h] = acc;
  }
  for (int i = t; i < 16 * 128; i += 256) {  // loop_sum[dst] += e
    int r = i >> 7, c = i & 127;
    atomicAdd(&loop_sum[(size_t)sDst[r] * 128 + c], sE[i]);
  }
  if (t < 16) atomicAdd(&deg[sDst[t]], 1.0f);
}

// self-loop logits: raw = lrelu(a_src[n]+a_dst[n] + (loop_sum[n]/deg) . V)
__global__ void k_loop_raw(const float* __restrict__ loop_sum,
                           const float* __restrict__ deg,
                           const float* __restrict__ V,
                           const float* __restrict__ a_src,
                           const float* __restrict__ a_dst,
                           float* __restrict__ alpha_l, float* __restrict__ m,
                           int H) {
  int i = blockIdx.x * blockDim.x + threadIdx.x;
  if (i >= NN * H) return;
  int n = i / H, h = i % H;
  float inv = 1.0f / fmaxf(deg[n], 1.0f);
  float acc = 0.f;
  for (int k = 0; k < 128; ++k) acc += loop_sum[(size_t)n * 128 + k] * V[k * H + h];
  float raw = leaky02(a_src[i] + a_dst[i] + acc * inv);
  alpha_l[i] = raw;
  atomicMaxF(&m[i], raw);
}

__global__ void k_edge_raw(const int* __restrict__ ei,
                           const float* __restrict__ a_src,
                           const float* __restrict__ a_dst,
                           float* __restrict__ a_edge, float* __restrict__ m,
                           int H) {
  long long i = (long long)blockIdx.x * blockDim.x + threadIdx.x;
  if (i >= (long long)NE * H) return;
  int e = (int)(i / H), h = (int)(i % H);
  int s = ei[e], d = ei[NE + e];
  float raw = leaky02(a_src[s * H + h] + a_dst[d * H + h] + a_edge[i]);
  a_edge[i] = raw;
  atomicMaxF(&m[d * H + h], raw);
}

__global__ void k_exp_edge(const int* __restrict__ ei, float* __restrict__ a_edge,
                           const float* __restrict__ m, float* __restrict__ ssum,
                           int H) {
  long long i = (long long)blockIdx.x * blockDim.x + threadIdx.x;
  if (i >= (long long)NE * H) return;
  int e = (int)(i / H), h = (int)(i % H);
  int d = ei[NE + e];
  float a = __expf(a_edge[i] - m[d * H + h]);
  a_edge[i] = a;
  atomicAdd(&ssum[d * H + h], a);
}

__global__ void k_exp_loop(float* __restrict__ alpha_l, const float* __restrict__ m,
                           float* __restrict__ ssum, int H) {
  int i = blockIdx.x * blockDim.x + threadIdx.x;
  if (i >= NN * H) return;
  float a = __expf(alpha_l[i] - m[i]);
  alpha_l[i] = a;
  atomicAdd(&ssum[i], a);
}

// out[d, :] += a_e * xp[src, :]  -- float4, no per-edge normalization (the
// softmax denominator is a per-(dst,head) scalar applied in k_finalize).
__global__ void k_agg_edge(const int* __restrict__ ei, const float* __restrict__ xp,
                           const float* __restrict__ a_edge,
                           float* __restrict__ out, int H, int C, int HC) {
  const int Q = HC >> 2;  // float4 groups per row (32 or 16)
  long long i = (long long)blockIdx.x * blockDim.x + threadIdx.x;
  if (i >= (long long)NE * Q) return;
  int e = (int)(i / Q), q = (int)(i % Q);
  int h = (q << 2) / C;
  int s = ei[e], d = ei[NE + e];
  float a = a_edge[(size_t)e * H + h];
  float4 v = *(const float4*)(xp + (size_t)s * HC + (q << 2));
  float* o = out + (size_t)d * HC + (q << 2);
  atomicAdd(o + 0, a * v.x);
  atomicAdd(o + 1, a * v.y);
  atomicAdd(o + 2, a * v.z);
  atomicAdd(o + 3, a * v.w);
}

__global__ void k_agg_loop(const float* __restrict__ xp,
                           const float* __restrict__ alpha_l,
                           float* __restrict__ out, int H, int C, int HC) {
  const int Q = HC >> 2;
  int i = blockIdx.x * blockDim.x + threadIdx.x;
  if (i >= NN * Q) return;
  int n = i / Q, q = i % Q;
  int h = (q << 2) / C;
  float a = alpha_l[n * H + h];
  float4 v = *(const float4*)(xp + (size_t)n * HC + (q << 2));
  float* o = out + (size_t)n * HC + (q << 2);
  atomicAdd(o + 0, a * v.x);
  atomicAdd(o + 1, a * v.y);
  atomicAdd(o + 2, a * v.z);
  atomicAdd(o + 3, a * v.w);
}

// dst = [relu]( out/(s+eps) + bias )   -- float4
__global__ void k_finalize(const float* __restrict__ out,
                           const float* __restrict__ ssum,
                           const float* __restrict__ bias, float* __restrict__ dst,
                           int H, int C, int HC, int do_relu) {
  const int Q = HC >> 2;
  int i = blockIdx.x * blockDim.x + threadIdx.x;
  if (i >= NN * Q) return;
  int n = i / Q, q = i % Q;
  int h = (q << 2) / C;
  float inv = 1.0f / (ssum[n * H + h] + 1e-16f);
  float4 v = *(const float4*)(out + (size_t)n * HC + (q << 2));
  float4 b = *(const float4*)(bias + (q << 2));
  float4 r;
  r.x = v.x * inv + b.x;
  r.y = v.y * inv + b.y;
  r.z = v.z * inv + b.z;
  r.w = v.w * inv + b.w;
  if (do_relu) {
    r.x = fmaxf(r.x, 0.f); r.y = fmaxf(r.y, 0.f);
    r.z = fmaxf(r.z, 0.f); r.w = fmaxf(r.w, 0.f);
  }
  *(float4*)(dst + (size_t)n * HC + (q << 2)) = r;
}

// ---------------------------------------------------------------------------
extern "C" void kernel_launch(void* const* d_in, const int* in_sizes, int n_in,
                              void* d_out, int out_size, void* d_ws, size_t ws_size,
                              hipStream_t stream) {
  (void)in_sizes; (void)n_in; (void)out_size; (void)ws_size;
  const float* x0 = (const float*)d_in[0];
  const int* ei = (const int*)d_in[1];      // edge_index [2,E] (int32 under JAX default)
  const float* ea = (const float*)d_in[2];  // edge_attr [E,32]
  // params per layer, insertion order: We1 be1 We2 be2 W_x W_e att_src att_dst att_edge bias
  auto P = [&](int l, int k) { return (const float*)d_in[3 + l * 10 + k]; };

  float* w = (float*)d_ws;
  float* xp = w;       w += (size_t)NN * 128;
  float* xbufA = w;    w += (size_t)NN * 128;
  float* xbufB = w;    w += (size_t)NN * 128;
  float* loop_sum = w; w += (size_t)NN * 128;
  float* outbuf = w;   w += (size_t)NN * 128;
  float* deg = w;      w += NN;
  float* a_src = w;    w += (size_t)NN * 4;
  float* a_dst = w;    w += (size_t)NN * 4;
  float* mbuf = w;     w += (size_t)NN * 4;
  float* sbuf = w;     w += (size_t)NN * 4;
  float* alpha_l = w;  w += (size_t)NN * 4;
  float* Vedge = w;    w += 128 * 4;
  float* a_edge = w;   w += (size_t)NE * 4;

  auto g1 = [](long long n) { return (unsigned)((n + 255) / 256); };
  const float* xin = x0;
  for (int l = 0; l < 3; ++l) {
    const int H = (l < 2) ? 4 : 1;
    const int C = (l < 2) ? 32 : 64;
    const int HC = H * C;
    const int Q = HC / 4;
    float* xout = (l == 0) ? xbufA : (l == 1) ? xbufB : (float*)d_out;

    k_fill<<<g1(NN * 128), 256, 0, stream>>>(loop_sum, NN * 128, 0.f);
    k_fill<<<g1(NN), 256, 0, stream>>>(deg, NN, 0.f);
    k_fill<<<g1(NN * H), 256, 0, stream>>>(sbuf, NN * H, 0.f);
    k_fill<<<g1(NN * H), 256, 0, stream>>>(mbuf, NN * H, -INFINITY);
    k_fill<<<g1((long long)NN * HC), 256, 0, stream>>>(outbuf, NN * HC, 0.f);

    k_prep_vedge<<<1, 128, 0, stream>>>(P(l, 5), P(l, 8), Vedge, H, C, HC);
    k_gemm_node<<<NN / 16, 256, 0, stream>>>(xin, P(l, 4), xp, HC);
    k_att_nodes<<<g1(NN * H), 256, 0, stream>>>(xp, P(l, 6), P(l, 7), a_src, a_dst, H, C, HC);
    k_edge_fused<<<NE / 16, 256, 0, stream>>>(ea, ei, P(l, 0), P(l, 1), P(l, 2),
                                              P(l, 3), Vedge, a_edge, loop_sum, deg, H);
    k_loop_raw<<<g1(NN * H), 256, 0, stream>>>(loop_sum, deg, Vedge, a_src, a_dst,
                                               alpha_l, mbuf, H);
    k_edge_raw<<<g1((long long)NE * H), 256, 0, stream>>>(ei, a_src, a_dst, a_edge, mbuf, H);
    k_exp_loop<<<g1(NN * H), 256, 0, stream>>>(alpha_l, mbuf, sbuf, H);
    k_exp_edge<<<g1((long long)NE * H), 256, 0, stream>>>(ei, a_edge, mbuf, sbuf, H);
    k_agg_edge<<<g1((long long)NE * Q), 256, 0, stream>>>(ei, xp, a_edge, outbuf, H, C, HC);
    k_agg_loop<<<g1((long long)NN * Q), 256, 0, stream>>>(xp, alpha_l, outbuf, H, C, HC);
    k_finalize<<<g1((long long)NN * Q), 256, 0, stream>>>(outbuf, sbuf, P(l, 9), xout,
                                                          H, C, HC, l < 2 ? 1 : 0);
    xin = xout;
  }
}